// SparseMatmulOnly_62878321214323
// MI455X (gfx1250) — compile-verified
//
#include <hip/hip_runtime.h>

// Problem constants from the reference
#define H_DIM 2048
#define E_DIM 8
#define T_DIM 2048
#define F_DIM 4096  // 2 * INTER

typedef __attribute__((ext_vector_type(2))) float v2f;
typedef __attribute__((ext_vector_type(4))) float v4f;
typedef __attribute__((ext_vector_type(8))) float v8f;

// ---------------------------------------------------------------------------
// Kernel A (bandwidth-dominant, ~256 MB streamed once):
//   wsum[e*H + h] = sum_f W[e,h,f]
// One 256-thread block per (e,h) row of F_DIM=4096 contiguous floats (16 KB).
// Non-temporal b128 loads: W (256 MB) exceeds L2 (192 MB), stream it.
// Block 0 also zeroes the scalar output (runs before kernel B in stream order).
// ---------------------------------------------------------------------------
__global__ void wsum_kernel(const float* __restrict__ W,
                            float* __restrict__ wsum,
                            float* __restrict__ out) {
  __shared__ float red[256];
  const int row = blockIdx.x;  // 0 .. E*H-1
  const v4f* p4 = (const v4f*)(W + (size_t)row * F_DIM);
  const int t = threadIdx.x;

  float s = 0.f;
#pragma unroll
  for (int j = 0; j < 4; ++j) {
    v4f v = __builtin_nontemporal_load(&p4[t + j * 256]);
    s += (v.x + v.y) + (v.z + v.w);
  }
  red[t] = s;
  __syncthreads();
#pragma unroll
  for (int off = 128; off > 0; off >>= 1) {
    if (t < off) red[t] += red[t + off];
    __syncthreads();
  }
  if (t == 0) {
    wsum[row] = red[0];
    if (row == 0) out[0] = 0.f;  // deterministic init each launch
  }
}

// ---------------------------------------------------------------------------
// Kernel B: Hm[e,h] = sum_t mask[e,t] * hidden[t,h] via V_WMMA_F32_16X16X4_F32
// (exact f32 matrix op), fused with the final dot against wsum.
//   A (16x4): rows m=0..15 are experts (m>=8 zero-padded), K = t-chunk of 4
//   B (4x16): hidden[t, h0..h0+15]
//   C/D (16x16) f32: lanes 0-15 hold M=0..7 in VGPRs 0..7 -> valid experts.
// 128 waves, one 16-wide h-tile each, K-loop over all T=2048 tokens.
// ---------------------------------------------------------------------------
__global__ void wmma_reduce_kernel(const float* __restrict__ hidden,
                                   const float* __restrict__ mask,
                                   const float* __restrict__ wsum,
                                   float* __restrict__ out) {
  const int lane = threadIdx.x & 31;
  const int wave = (blockIdx.x * blockDim.x + threadIdx.x) >> 5;  // 0..127
  const int half = lane >> 4;   // 0: K=0,1 / M rows 0..7 ; 1: K=2,3 / M rows 8..15
  const int nrow = lane & 15;   // A: M index, B/C/D: N index
  const int h0   = wave * 16;

  // Zero-pad expert rows >= E without divergent loads (keep EXEC uniform for WMMA)
  const float ascale = (nrow < E_DIM) ? 1.f : 0.f;
  const float* arow_p = mask + (size_t)((nrow < E_DIM) ? nrow : 0) * T_DIM;

  v8f acc = {};
  for (int t0 = 0; t0 < T_DIM; t0 += 4) {
    const int k = t0 + half * 2;  // this half-wave's K pair
    v2f av, bv;
    av[0] = arow_p[k]     * ascale;                 // A[m, k]
    av[1] = arow_p[k + 1] * ascale;                 // A[m, k+1]
    bv[0] = hidden[(size_t)(k)     * H_DIM + h0 + nrow];  // B[k,   n]
    bv[1] = hidden[(size_t)(k + 1) * H_DIM + h0 + nrow];  // B[k+1, n]
    // 8 args: (neg_a, A, neg_b, B, c_mod, C, reuse_a, reuse_b)
    acc = __builtin_amdgcn_wmma_f32_16x16x4_f32(
        false, av, false, bv, (short)0, acc, false, false);
  }

  // Epilogue: lanes 0-15 hold D[M=r, N=nrow] = Hm[e=r, h0+nrow] in acc[r]
  if (half == 0) {
    const int h = h0 + nrow;
    float part = 0.f;
#pragma unroll
    for (int r = 0; r < 8; ++r) part += acc[r] * wsum[r * H_DIM + h];
    atomicAdd(out, part);
  }
}

// ---------------------------------------------------------------------------
extern "C" void kernel_launch(void* const* d_in, const int* in_sizes, int n_in,
                              void* d_out, int out_size, void* d_ws, size_t ws_size,
                              hipStream_t stream) {
  (void)in_sizes; (void)n_in; (void)out_size; (void)ws_size;
  const float* hidden = (const float*)d_in[0];  // [1,1,T,H]  f32
  const float* W      = (const float*)d_in[1];  // [1,E,H,2*INTER] f32
  const float* mask   = (const float*)d_in[2];  // [1,E,T,1]  f32
  float* out  = (float*)d_out;                  // scalar
  float* wsum = (float*)d_ws;                   // E*H floats = 64 KB scratch

  // A: stream-reduce W over f (HBM-bound, ~11 us at 23.3 TB/s) + zero out
  wsum_kernel<<<E_DIM * H_DIM, 256, 0, stream>>>(W, wsum, out);

  // B: masked-token matmul via f32 WMMA + fused final dot-reduce
  const int waves  = H_DIM / 16;          // 128 h-tiles
  const int blocks = waves * 32 / 256;    // 16 blocks of 8 waves
  wmma_reduce_kernel<<<blocks, 256, 0, stream>>>(hidden, mask, wsum, out);
}